// MinibatchDiscrimination_68582037782886
// MI455X (gfx1250) — compile-verified
//
#include <hip/hip_runtime.h>
#include <hip/hip_bf16.h>
#include <cmath>

// ---------------------------------------------------------------------------
// MinibatchDiscrimination, fused single kernel for gfx1250 (MI455X).
//   M = X @ T              via v_wmma_f32_16x16x4_f32 (f32 WMMA, K-step 4)
//   feats via pairwise L1 + exp, computed out of LDS (M never touches HBM)
//   out = concat(x, feats)
// ---------------------------------------------------------------------------

typedef __attribute__((ext_vector_type(2))) float v2f;
typedef __attribute__((ext_vector_type(8))) float v8f;

#define F_DIM     256   // input feature dim
#define KD_DIM    512   // NUM_KERNELS * KERNEL_DIM
#define NK        32    // NUM_KERNELS
#define KDIM      16    // KERNEL_DIM
#define TILE_N    16    // samples per workgroup
#define NTHREADS  512   // 16 waves (wave32)
#define XS_STRIDE 260   // 256 + 4 pad  -> conflict-free A-frag column reads
#define MS_STRIDE 516   // 512 + 4 pad  -> conflict-free C-tile scatter

__global__ __launch_bounds__(NTHREADS)
void mbdisc_fused_kernel(const float* __restrict__ x,
                         const float* __restrict__ T,
                         float* __restrict__ out) {
  __shared__ float xs[TILE_N * XS_STRIDE];  // 16.25 KB : X tile [16 x 256]
  __shared__ float ms[TILE_N * MS_STRIDE];  // 32.25 KB : M tile [16 x 512]

  const int tid  = threadIdx.x;
  const int wave = tid >> 5;
  const int lane = tid & 31;
  const int n0   = blockIdx.x * TILE_N;

  // -------- stage 0: stage X tile into LDS (float4, fully coalesced) -------
  {
    const float4* xg = reinterpret_cast<const float4*>(x + (size_t)n0 * F_DIM);
#pragma unroll
    for (int i = 0; i < 2; ++i) {
      int idx = tid + i * NTHREADS;        // 0..1023 over 16*64 float4s
      int r   = idx >> 6;
      int c4  = idx & 63;
      float4 v = xg[r * (F_DIM / 4) + c4];
      float* dst = &xs[r * XS_STRIDE + c4 * 4];
      dst[0] = v.x; dst[1] = v.y; dst[2] = v.z; dst[3] = v.w;
    }
  }
  __syncthreads();

  // -------- stage 1: M_tile[16,512] = X_tile[16,256] @ T[256,512] ----------
  // wave w computes column tiles 2w and 2w+1 (16 cols each).
  // A frag (16x4 f32): lanes 0-15 -> {K0,K1}, lanes 16-31 -> {K2,K3}, M=lane%16
  // B frag (4x16 f32): same K split across half-waves, N=lane%16
  // C/D (16x16 f32):  VGPR r -> row r (lanes 0-15) / row r+8 (lanes 16-31)
  {
    const int row = lane & 15;
    const int kh  = (lane >> 4) << 1;  // 0 or 2
    const int hi  = lane >> 4;         // 0 or 1
#pragma unroll
    for (int t = 0; t < 2; ++t) {
      const int col = (wave * 2 + t) * 16;
      const float* tp = T + col + (lane & 15);
      v8f c = {0.f, 0.f, 0.f, 0.f, 0.f, 0.f, 0.f, 0.f};
#pragma unroll 8
      for (int k = 0; k < F_DIM; k += 4) {
        v2f a = *reinterpret_cast<const v2f*>(&xs[row * XS_STRIDE + k + kh]);
        v2f b;
        b.x = tp[(size_t)(k + kh + 0) * KD_DIM];
        b.y = tp[(size_t)(k + kh + 1) * KD_DIM];
        c = __builtin_amdgcn_wmma_f32_16x16x4_f32(
            /*neg_a=*/false, a, /*neg_b=*/false, b,
            /*c_mod=*/(short)0, c, /*reuse_a=*/false, /*reuse_b=*/false);
      }
#pragma unroll
      for (int r = 0; r < 8; ++r) {
        ms[(r + hi * 8) * MS_STRIDE + col + (lane & 15)] = c[r];
      }
    }
  }
  __syncthreads();

  // -------- stage 2: pairwise L1 across kernels + exp ----------------------
  // wave w handles sample n=w; lane = k1.
  {
    const int n  = wave;
    const int k1 = lane;
    const float* mrow = &ms[n * MS_STRIDE];

    float4 m1v[4], acc[4];
#pragma unroll
    for (int d4 = 0; d4 < 4; ++d4) {
      m1v[d4] = *reinterpret_cast<const float4*>(&mrow[k1 * KDIM + d4 * 4]);
      acc[d4] = make_float4(0.f, 0.f, 0.f, 0.f);
    }
#pragma unroll 4
    for (int k2 = 0; k2 < NK; ++k2) {
#pragma unroll
      for (int d4 = 0; d4 < 4; ++d4) {
        // broadcast read: all lanes in the wave read the same LDS address
        float4 m2 = *reinterpret_cast<const float4*>(&mrow[k2 * KDIM + d4 * 4]);
        acc[d4].x += fabsf(m2.x - m1v[d4].x);
        acc[d4].y += fabsf(m2.y - m1v[d4].y);
        acc[d4].z += fabsf(m2.z - m1v[d4].z);
        acc[d4].w += fabsf(m2.w - m1v[d4].w);
      }
    }
    float feat = 0.f;
#pragma unroll
    for (int d4 = 0; d4 < 4; ++d4) {
      feat += __expf(-acc[d4].x) + __expf(-acc[d4].y) +
              __expf(-acc[d4].z) + __expf(-acc[d4].w);
    }
    out[(size_t)(n0 + n) * (F_DIM + NK) + F_DIM + k1] = feat;
  }

  // -------- stage 3: copy x rows into out (from LDS, float4 stores) --------
  {
#pragma unroll
    for (int i = 0; i < 2; ++i) {
      int idx = tid + i * NTHREADS;
      int r   = idx >> 6;
      int c4  = idx & 63;
      const float* src = &xs[r * XS_STRIDE + c4 * 4];
      float4 v = make_float4(src[0], src[1], src[2], src[3]);
      *reinterpret_cast<float4*>(
          &out[(size_t)(n0 + r) * (F_DIM + NK) + c4 * 4]) = v;
    }
  }
}

extern "C" void kernel_launch(void* const* d_in, const int* in_sizes, int n_in,
                              void* d_out, int out_size, void* d_ws, size_t ws_size,
                              hipStream_t stream) {
  (void)n_in; (void)out_size; (void)d_ws; (void)ws_size;
  const float* x = (const float*)d_in[0];   // [B,S,256] f32
  const float* T = (const float*)d_in[1];   // [256,512] f32
  float* out = (float*)d_out;               // [B,S,288] f32

  const int N = in_sizes[0] / F_DIM;        // B*S = 16384
  const int blocks = N / TILE_N;            // 1024 workgroups
  mbdisc_fused_kernel<<<blocks, NTHREADS, 0, stream>>>(x, T, out);
}